// GraphAttentionLayer_30683246362649
// MI455X (gfx1250) — compile-verified
//
#include <hip/hip_runtime.h>
#include <math.h>

typedef __attribute__((ext_vector_type(2))) float v2f;
typedef __attribute__((ext_vector_type(8))) float v8f;

#define NEG_SLOPE 0.2f

static constexpr int kB = 8;
static constexpr int kT = 2048;
static constexpr int kD = 256;
static constexpr int kM = kB * kT;       // 16384 rows of h
static constexpr int XS_STRIDE = 260;    // LDS pad: A-pattern (4m+k) conflict-free
static constexpr int HS_STRIDE = 264;    // LDS pad: B-pattern half-waves on disjoint banks
static constexpr int PS_STRIDE = 17;     // LDS pad: 17m+k distinct mod 64

// single merged LDS arena for k_attn (lone __shared__ object -> LDS offset 0,
// so async-to-LDS instructions can take literal byte addresses)
static constexpr int HS_BUF   = 16 * HS_STRIDE;          // 4224 floats / buffer
static constexpr int PS_BUF   = 16 * PS_STRIDE;          // 272 floats / buffer
static constexpr int SM_HS    = 0;                       // hs[2]
static constexpr int SM_PS    = SM_HS + 2 * HS_BUF;      // ps[2]
static constexpr int SM_LRED  = SM_PS + 2 * PS_BUF;      // lred
static constexpr int SM_S1    = SM_LRED + PS_BUF;        // s1s[16]
static constexpr int SM_SM    = SM_S1 + 16;              // sm[16]
static constexpr int SM_LINV  = SM_SM + 16;              // linv[16]
static constexpr int SM_TOTAL = SM_LINV + 16;            // 9312 floats (~36.4 KB)

__device__ __forceinline__ float lrelu(float e) {
  return e > 0.0f ? e : NEG_SLOPE * e;
}

// Async DMA: 16 bytes global -> LDS, bypassing VGPRs (ASYNCcnt-tracked).
// GVS mode: mem = SADDR(64) + VADDR(32) ; VDST = per-lane LDS byte address.
__device__ __forceinline__ void async_b128_to_lds(unsigned lds_byte,
                                                  unsigned gbl_off,
                                                  const float* base) {
  asm volatile("global_load_async_to_lds_b128 %0, %1, %2"
               :
               : "v"(lds_byte), "v"(gbl_off), "s"(base)
               : "memory");
}
__device__ __forceinline__ void wait_async0() {
  asm volatile("s_wait_asynccnt 0x0" ::: "memory");
}

// ---------------------------------------------------------------------------
// Kernel 1: h[m,n] = sum_d x[m,d] * W[n,d]   (M=16384, N=256, K=256), f32 WMMA.
// Block = 16-row M tile staged in LDS; 8 waves x two 16-col N tiles each.
// ---------------------------------------------------------------------------
__global__ __launch_bounds__(256)
void k_h_gemm(const float* __restrict__ x, const float* __restrict__ W,
              float* __restrict__ h) {
  __shared__ __align__(16) float xs[16 * XS_STRIDE];
  const int tid = threadIdx.x;
  const int m0 = blockIdx.x * 16;

  { // stage 16x256 x-tile (float4, 16 threads per row)
    const int r = tid >> 4, tc = tid & 15;
    const float4* x4 = (const float4*)x;
    #pragma unroll
    for (int i = 0; i < 4; ++i) {
      float4 v = x4[(size_t)(m0 + r) * (kD / 4) + tc + i * 16];
      *(float4*)&xs[r * XS_STRIDE + (tc + i * 16) * 4] = v;
    }
  }
  __syncthreads();

  const int wave = tid >> 5, lane = tid & 31;
  const int mrow  = lane & 15;
  const int khalf = (lane >> 4) * 2;

  #pragma unroll
  for (int t = 0; t < 2; ++t) {
    const int n = (wave * 2 + t) * 16 + (lane & 15);
    v8f c = {};
    #pragma unroll 4
    for (int k0 = 0; k0 < kD; k0 += 4) {
      const int k = k0 + khalf;
      v2f a = *(const v2f*)&xs[mrow * XS_STRIDE + k];
      v2f b = *(const v2f*)&W[(size_t)n * kD + k];   // B[k][n] = W[n][k]
      c = __builtin_amdgcn_wmma_f32_16x16x4_f32(false, a, false, b, (short)0, c,
                                                false, false);
    }
    #pragma unroll
    for (int r = 0; r < 8; ++r) {
      const int m = m0 + r + 8 * (lane >> 4);
      h[(size_t)m * kD + (wave * 2 + t) * 16 + (lane & 15)] = c[r];
    }
  }
}

// ---------------------------------------------------------------------------
// Kernel 2: s1 = h @ a[:256], s2 = h @ a[256:]
// ---------------------------------------------------------------------------
__global__ __launch_bounds__(128)
void k_scores(const float* __restrict__ h, const float* __restrict__ a,
              float* __restrict__ s1, float* __restrict__ s2) {
  const int row = blockIdx.x * 128 + threadIdx.x;
  const float4* hr = (const float4*)(h + (size_t)row * kD);
  const float4* a1 = (const float4*)a;
  const float4* a2 = (const float4*)(a + kD);
  float acc1 = 0.0f, acc2 = 0.0f;
  #pragma unroll 4
  for (int i = 0; i < kD / 4; ++i) {
    float4 hv = hr[i], v1 = a1[i], v2 = a2[i];
    acc1 += hv.x * v1.x + hv.y * v1.y + hv.z * v1.z + hv.w * v1.w;
    acc2 += hv.x * v2.x + hv.y * v2.y + hv.z * v2.z + hv.w * v2.w;
  }
  s1[row] = acc1;
  s2[row] = acc2;
}

// ---------------------------------------------------------------------------
// Kernel 3: m2[b] = max_j s2[b,j]  (lrelu monotone => exact softmax row max
// is lrelu(s1_i + m2[b]); no online rescaling needed in the attention pass)
// ---------------------------------------------------------------------------
__global__ __launch_bounds__(256)
void k_colmax(const float* __restrict__ s2, float* __restrict__ m2) {
  __shared__ float red[256];
  const int b = blockIdx.x, tid = threadIdx.x;
  float m = -3.402823466e38f;
  for (int j = tid; j < kT; j += 256) m = fmaxf(m, s2[b * kT + j]);
  red[tid] = m;
  __syncthreads();
  for (int s = 128; s > 0; s >>= 1) {
    if (tid < s) red[tid] = fmaxf(red[tid], red[tid + s]);
    __syncthreads();
  }
  if (tid == 0) m2[b] = red[0];
}

// ---------------------------------------------------------------------------
// Kernel 4: one-pass attention + ELU, double-buffered with async-to-LDS DMA.
// Per iteration: prefetch next 16x256 h-tile (DMA) + next p-tile into buffer
// 1-cur while the waves run 8 f32 WMMAs each on buffer cur; then
// s_wait_asynccnt 0 + one barrier.
// ---------------------------------------------------------------------------
__global__ __launch_bounds__(256)
void k_attn(const float* __restrict__ h, const float* __restrict__ s1,
            const float* __restrict__ s2, const float* __restrict__ m2,
            float* __restrict__ out) {
  __shared__ __align__(16) float smem[SM_TOTAL];

  const int tid = threadIdx.x;
  const int i0  = blockIdx.x * 16;   // query tile
  const int b   = blockIdx.y;        // batch

  if (tid < 16) {
    float v = s1[b * kT + i0 + tid];
    smem[SM_S1 + tid] = v;
    smem[SM_SM + tid] = lrelu(v + m2[b]);  // exact row max of logits
  }
  __syncthreads();

  const int wave = tid >> 5, lane = tid & 31;
  const int pi = tid & 15, pj = tid >> 4;  // p-tile coords; also (tc, jr) for DMA
  const int khalf = (lane >> 4) * 2;
  const int n0 = (wave * 2 + 0) * 16 + (lane & 15);
  const int n1 = (wave * 2 + 1) * 16 + (lane & 15);
  const float mi  = smem[SM_SM + pi];
  const float s1i = smem[SM_S1 + pi];

  v8f c0 = {}, c1 = {};
  float lpart = 0.0f;

  // stage(j0, buf): p-tile (VALU+DS) and h-tile (async DMA) into buffer buf
  auto stage = [&](int j0, int buf) {
    float e = lrelu(s1i + s2[b * kT + j0 + pj]);
    float p = __expf(e - mi);
    smem[SM_PS + buf * PS_BUF + pi * PS_STRIDE + pj] = p;
    lpart += p;
    const unsigned row = (unsigned)(b * kT + j0 + pj);   // jr = pj
    #pragma unroll
    for (int i = 0; i < 4; ++i) {
      const unsigned col16 = (unsigned)(pi + i * 16);    // tc = pi
      const unsigned ldsb = (unsigned)((SM_HS + buf * HS_BUF + pj * HS_STRIDE) * 4
                                       + col16 * 16);
      const unsigned goff = row * (unsigned)(kD * 4) + col16 * 16;
      async_b128_to_lds(ldsb, goff, h);
    }
  };

  stage(0, 0);
  wait_async0();
  __syncthreads();

  for (int jt = 0; jt < kT / 16; ++jt) {
    const int cur = jt & 1;
    if (jt + 1 < kT / 16) stage((jt + 1) * 16, cur ^ 1);  // prefetch overlaps WMMA

    const float* hsb = &smem[SM_HS + cur * HS_BUF];
    const float* psb = &smem[SM_PS + cur * PS_BUF];
    #pragma unroll
    for (int k0 = 0; k0 < 16; k0 += 4) {
      const int k = k0 + khalf;
      v2f a;
      a.x = psb[(lane & 15) * PS_STRIDE + k];
      a.y = psb[(lane & 15) * PS_STRIDE + k + 1];
      v2f bf0; bf0.x = hsb[k * HS_STRIDE + n0]; bf0.y = hsb[(k + 1) * HS_STRIDE + n0];
      c0 = __builtin_amdgcn_wmma_f32_16x16x4_f32(false, a, false, bf0, (short)0,
                                                 c0, false, false);
      v2f bf1; bf1.x = hsb[k * HS_STRIDE + n1]; bf1.y = hsb[(k + 1) * HS_STRIDE + n1];
      c1 = __builtin_amdgcn_wmma_f32_16x16x4_f32(false, a, false, bf1, (short)0,
                                                 c1, false, false);
    }
    wait_async0();       // this wave's DMAs into buf 1-cur are in LDS
    __syncthreads();     // everyone's DMAs + p stores visible; cur reads done
  }

  // row sums l_i: 16 partials per row
  smem[SM_LRED + pi * PS_STRIDE + pj] = lpart;
  __syncthreads();
  if (tid < 16) {
    float l = 0.0f;
    #pragma unroll
    for (int j = 0; j < 16; ++j) l += smem[SM_LRED + tid * PS_STRIDE + j];
    smem[SM_LINV + tid] = 1.0f / l;   // l >= 1 (max term contributes exp(0)=1)
  }
  __syncthreads();

  #pragma unroll
  for (int t = 0; t < 2; ++t) {
    const v8f c = t ? c1 : c0;
    const int n = t ? n1 : n0;
    #pragma unroll
    for (int r = 0; r < 8; ++r) {
      const int m = r + 8 * (lane >> 4);
      float v = c[r] * smem[SM_LINV + m];
      v = v > 0.0f ? v : (__expf(v) - 1.0f);   // ELU
      out[((size_t)(b * kT) + i0 + m) * kD + n] = v;
    }
  }
}

// ---------------------------------------------------------------------------
extern "C" void kernel_launch(void* const* d_in, const int* in_sizes, int n_in,
                              void* d_out, int out_size, void* d_ws, size_t ws_size,
                              hipStream_t stream) {
  (void)in_sizes; (void)n_in; (void)out_size; (void)ws_size;
  const float* x = (const float*)d_in[0];   // (8, 2048, 256)
  const float* W = (const float*)d_in[1];   // (256, 256)
  const float* a = (const float*)d_in[2];   // (512,)
  float* out = (float*)d_out;               // (8, 2048, 256)

  // workspace layout: h (16 MB) | s1 | s2 | m2   (~16.9 MB total)
  float* h  = (float*)d_ws;
  float* s1 = h + (size_t)kM * kD;
  float* s2 = s1 + kM;
  float* m2 = s2 + kM;

  k_h_gemm<<<kM / 16, 256, 0, stream>>>(x, W, h);
  k_scores<<<kM / 128, 128, 0, stream>>>(h, a, s1, s2);
  k_colmax<<<kB, 256, 0, stream>>>(s2, m2);
  dim3 g2(kT / 16, kB);
  k_attn<<<g2, 256, 0, stream>>>(h, s1, s2, m2, out);
}